// RNN_83614423318549
// MI455X (gfx1250) — compile-verified
//
#include <hip/hip_runtime.h>

// RNN scan fused kernel for MI455X (gfx1250, wave32).
//   h_t = gelu(h_{t-1} @ Wh^T + x_t @ Win^T),  H=128, B=128, SEQ=4096, 2 layers
// Grid: (8 batch tiles of 16 rows) x (2 layers). Block: 256 threads = 8 waves.
// Wave w owns output columns [16w, 16w+16). Weight B-fragments live in VGPRs
// (reused 4096x). h and x tiles double-buffer through bank-padded LDS.
// Per-step sync uses the CDNA5 split barrier: signal early after LDS stores,
// overlap global state-stores + next-x prefetch with the barrier wait.

typedef __attribute__((ext_vector_type(2))) float v2f;
typedef __attribute__((ext_vector_type(4))) float v4f;
typedef __attribute__((ext_vector_type(8))) float v8f;

#define SEQ   4096
#define BATCH 128
#define HID   128
#define LSTR  132   // 128 + 4 floats pad: ds_load_b64 A-frags hit all 64 banks

__global__ __launch_bounds__(256)
void rnn_fused_kernel(const float* __restrict__ seq,
                      const float* __restrict__ init_state,
                      const float* __restrict__ Win0,
                      const float* __restrict__ Wh0,
                      const float* __restrict__ Win1,
                      const float* __restrict__ Wh1,
                      float* __restrict__ out)
{
    __shared__ float xbuf[2][16 * LSTR];
    __shared__ float hbuf[2][16 * LSTR];

    const int tid   = threadIdx.x;
    const int wave  = tid >> 5;
    const int lane  = tid & 31;
    const int l15   = lane & 15;
    const int khalf = lane >> 4;          // 0: lanes 0-15, 1: lanes 16-31
    const int layer = blockIdx.y;
    const int b0    = blockIdx.x * 16;    // batch-tile row base (M)
    const int n0    = wave * 16;          // output-column base (N)

    const float* Win = layer ? Win1 : Win0;
    const float* Wh  = layer ? Wh1  : Wh0;
    const bool write_states = (layer == 1);

    float* out_states = out;                                 // (SEQ,B,H)
    float* out_finals = out + (size_t)SEQ * BATCH * HID;     // (2*B,H)

    // ---- Pin B-operand weight fragments in VGPRs (K=128 -> 32 chunks) ----
    // B(k,n) = W[n][k]; chunk kk: VGPR0 = W[n][4kk+2*khalf], VGPR1 = +1.
    v2f wfin[32], wfh[32];
    {
        const int n = n0 + l15;
        #pragma unroll
        for (int kk = 0; kk < 32; ++kk) {
            const int k = 4 * kk + 2 * khalf;
            wfin[kk] = *(const v2f*)(Win + n * HID + k);
            wfh[kk]  = *(const v2f*)(Wh  + n * HID + k);
        }
    }

    // ---- Prologue: stage x_0 -> xbuf[0]; init_state -> hbuf[1]; prefetch x_1
    const int r = tid >> 4;               // 0..15 (tile row)
    const int c = (tid & 15) * 8;         // 0..120 (8 floats per thread)
    v4f p0 = {}, p1 = {};                 // in-flight x_{t+1} tile fragment
    {
        const float* s = seq + ((size_t)0 * BATCH + b0 + r) * HID + c;
        v4f a0 = *(const v4f*)(s);
        v4f a1 = *(const v4f*)(s + 4);
        *(v4f*)(&xbuf[0][r * LSTR + c])     = a0;
        *(v4f*)(&xbuf[0][r * LSTR + c + 4]) = a1;
        const float* hs = init_state + layer * HID + c;      // (2,1,H), bcast over B
        v4f h0 = *(const v4f*)(hs);
        v4f h1 = *(const v4f*)(hs + 4);
        *(v4f*)(&hbuf[1][r * LSTR + c])     = h0;
        *(v4f*)(&hbuf[1][r * LSTR + c + 4]) = h1;
        const float* s1 = seq + ((size_t)1 * BATCH + b0 + r) * HID + c;
        p0 = *(const v4f*)(s1);
        p1 = *(const v4f*)(s1 + 4);
    }
    __syncthreads();

    v8f dreg = {};
    for (int t = 0; t < SEQ; ++t) {
        const float* xb = &xbuf[t & 1][0];
        const float* hb = &hbuf[(t + 1) & 1][0];   // h_{t-1} (parity (t-1)&1)

        // 4 accumulators: dependent-WMMA chain length 16 instead of 64
        v8f cx0 = {}, cx1 = {}, ch0 = {}, ch1 = {};

        // x_t @ Win^T  (independent of h -> fills latency of the h chain)
        #pragma unroll
        for (int kk = 0; kk < 32; kk += 2) {
            v2f a0 = *(const v2f*)(xb + l15 * LSTR + 4 * kk + 2 * khalf);
            cx0 = __builtin_amdgcn_wmma_f32_16x16x4_f32(
                      false, a0, false, wfin[kk], (short)0, cx0, false, false);
            v2f a1 = *(const v2f*)(xb + l15 * LSTR + 4 * (kk + 1) + 2 * khalf);
            cx1 = __builtin_amdgcn_wmma_f32_16x16x4_f32(
                      false, a1, false, wfin[kk + 1], (short)0, cx1, false, false);
        }
        // h_{t-1} @ Wh^T  (critical path)
        #pragma unroll
        for (int kk = 0; kk < 32; kk += 2) {
            v2f a0 = *(const v2f*)(hb + l15 * LSTR + 4 * kk + 2 * khalf);
            ch0 = __builtin_amdgcn_wmma_f32_16x16x4_f32(
                      false, a0, false, wfh[kk], (short)0, ch0, false, false);
            v2f a1 = *(const v2f*)(hb + l15 * LSTR + 4 * (kk + 1) + 2 * khalf);
            ch1 = __builtin_amdgcn_wmma_f32_16x16x4_f32(
                      false, a1, false, wfh[kk + 1], (short)0, ch1, false, false);
        }

        // exact GELU: 0.5*v*(1+erf(v/sqrt(2)))
        #pragma unroll
        for (int i = 0; i < 8; ++i) {
            float v = (cx0[i] + cx1[i]) + (ch0[i] + ch1[i]);
            dreg[i] = 0.5f * v * (1.0f + erff(v * 0.70710678118654752440f));
        }

        // h_t -> LDS (C/D layout: lane half selects M=i or M=i+8), conflict-free
        float* hw = &hbuf[t & 1][0];
        #pragma unroll
        for (int i = 0; i < 8; ++i) {
            const int m = i + 8 * khalf;
            hw[m * LSTR + n0 + l15] = dreg[i];
        }
        // commit prefetched x_{t+1} into the other LDS buffer (before signal!)
        if (t + 1 < SEQ) {
            float* xw = &xbuf[(t + 1) & 1][0];
            *(v4f*)(&xw[r * LSTR + c])     = p0;
            *(v4f*)(&xw[r * LSTR + c + 4]) = p1;
        }

        // ---- split barrier: make LDS visible, announce arrival early ----
        asm volatile("s_wait_dscnt 0x0\n\t"
                     "s_barrier_signal -1" ::: "memory");

        // Overlap with sibling waves' arrival: stream states + issue prefetch.
        if (write_states) {
            #pragma unroll
            for (int i = 0; i < 8; ++i) {
                const int m = i + 8 * khalf;
                out_states[((size_t)t * BATCH + b0 + m) * HID + n0 + l15] = dreg[i];
            }
        }
        if (t + 2 < SEQ) {      // prefetch x_{t+2}; completes under next step
            const float* s = seq + ((size_t)(t + 2) * BATCH + b0 + r) * HID + c;
            p0 = *(const v4f*)(s);
            p1 = *(const v4f*)(s + 4);
        }

        asm volatile("s_barrier_wait -1" ::: "memory");
    }

    // final hiddens: concat(layer0_final, layer1_final) -> (2*B, H)
    #pragma unroll
    for (int i = 0; i < 8; ++i) {
        const int m = i + 8 * khalf;
        out_finals[((size_t)layer * BATCH + b0 + m) * HID + n0 + l15] = dreg[i];
    }
}

extern "C" void kernel_launch(void* const* d_in, const int* in_sizes, int n_in,
                              void* d_out, int out_size, void* d_ws, size_t ws_size,
                              hipStream_t stream) {
    (void)in_sizes; (void)n_in; (void)out_size; (void)d_ws; (void)ws_size;
    const float* seq  = (const float*)d_in[0];
    const float* init = (const float*)d_in[1];
    const float* Win0 = (const float*)d_in[2];
    const float* Wh0  = (const float*)d_in[3];
    const float* Win1 = (const float*)d_in[4];
    const float* Wh1  = (const float*)d_in[5];
    float* out = (float*)d_out;

    dim3 grid(BATCH / 16, 2);   // 8 batch tiles x 2 independent layers
    dim3 block(256);            // 8 waves: one 16-col N-tile per wave
    rnn_fused_kernel<<<grid, block, 0, stream>>>(seq, init, Win0, Wh0, Win1, Wh1, out);
}